// CausalMultiheadSelfAttention_6038724018418
// MI455X (gfx1250) — compile-verified
//
#include <hip/hip_runtime.h>
#include <math.h>

typedef __attribute__((ext_vector_type(2))) float v2f;
typedef __attribute__((ext_vector_type(8))) float v8f;
typedef __attribute__((ext_vector_type(4))) unsigned int v4u;
typedef __attribute__((ext_vector_type(8))) int v8i;
typedef __attribute__((ext_vector_type(4))) int v4i;

#define D_MODEL 1024
#define NHEAD   16
#define DK      64
#define SEQ     2048
#define BSZ     2

#define KCHUNK  128          // K elements staged per TDM transfer
#define KPITCH  130          // LDS row pitch in floats (TDM pad: +2 dwords/row)
#define NCHUNK  (D_MODEL / KCHUNK)

__device__ __forceinline__ v8f wmma4(v2f a, v2f b, v8f c) {
  // D = A(16x4,f32) * B(4x16,f32) + C(16x16,f32)  -> v_wmma_f32_16x16x4_f32
  return __builtin_amdgcn_wmma_f32_16x16x4_f32(
      false, a, false, b, (short)0, c, false, false);
}

// ---------------------------------------------------------------------------
// TDM: DMA a 64-row x 128-col f32 tile of W (row stride D_MODEL) into LDS at
// byte offset `laddr`, with +2 dword padding after every 128 dwords so the
// LDS row pitch becomes KPITCH=130 floats (bank-conflict mitigation).
// D# layout per CDNA5 ISA. Tracked by TENSORcnt.
// ---------------------------------------------------------------------------
__device__ __forceinline__ void tdm_load_panel(const float* W, int n0, int c,
                                               unsigned int laddr) {
  unsigned long long ga =
      (unsigned long long)(size_t)W +
      ((size_t)n0 * D_MODEL + (size_t)c * KCHUNK) * 4ull;
  v4u g0;
  g0[0] = 1u;                                    // count=1, user-mode
  g0[1] = laddr;                                 // lds_addr (bytes)
  g0[2] = (unsigned int)ga;                      // global_addr[31:0]
  g0[3] = (unsigned int)((ga >> 32) & 0x1FFFFFFull) | (2u << 30); // [56:32]|type=2
  v8i g1;
  g1[0] = (int)((2u << 16)        // data_size = 4B
              | (1u << 20)        // pad_enable
              | (6u << 22)        // pad_interval: 128 dwords
              | (1u << 25));      // pad_amount: 2 dwords
  g1[1] = (int)((unsigned)D_MODEL << 16);   // tensor_dim0[15:0] in [31:16]
  g1[2] = (int)((unsigned)D_MODEL << 16);   // dim0 hi=0 | tensor_dim1[15:0]
  g1[3] = (int)((unsigned)KCHUNK << 16);    // dim1 hi=0 | tile_dim0=128
  g1[4] = 64;                               // tile_dim1=64 rows, tile_dim2=0
  g1[5] = D_MODEL;                          // tensor_dim0_stride lo32
  g1[6] = 0;                                // stride hi | dim1_stride lo
  g1[7] = 0;
  v4i gz4 = {0, 0, 0, 0};
  v8i gz8 = {0, 0, 0, 0, 0, 0, 0, 0};
  __builtin_amdgcn_tensor_load_to_lds(g0, g1, gz4, gz4, gz8, 0);
}

// ---------------------------------------------------------------------------
// Shared GEMM core: block computes C[128 x 64] of  C = A @ W^T.
// Wave `wid` owns a 32(M) x 64(N) tile: 2 A-frags x 4 B-frags -> 8 WMMA/step.
// W panel (64 rows x KCHUNK) double-buffered in LDS via TDM; wave 0 drives DMA.
// ---------------------------------------------------------------------------
__device__ __forceinline__ void gemm_core(const float* __restrict__ A,
                                          const float* __restrict__ W,
                                          float (*shw)[64 * KPITCH],
                                          int m0w, int n0,
                                          int wid, int lane, v8f acc[8]) {
  const int hi = lane >> 4, lo = lane & 15;
  const float* arow0 = A + (size_t)(m0w + lo) * D_MODEL + 2 * hi;
  const float* arow1 = A + (size_t)(m0w + 16 + lo) * D_MODEL + 2 * hi;
  const unsigned int lds0 = (unsigned int)(size_t)(void*)&shw[0][0];
  const unsigned int lds1 = (unsigned int)(size_t)(void*)&shw[1][0];

  if (wid == 0) tdm_load_panel(W, n0, 0, lds0);

  for (int c = 0; c < NCHUNK; ++c) {
    if (wid == 0) {
      if (c + 1 < NCHUNK) {
        tdm_load_panel(W, n0, c + 1, (c + 1) & 1 ? lds1 : lds0);
        __builtin_amdgcn_s_wait_tensorcnt(1);   // chunk c complete (in-order)
      } else {
        __builtin_amdgcn_s_wait_tensorcnt(0);
      }
    }
    __syncthreads();                            // publish LDS tile to all waves

    const float* shp = &shw[c & 1][0];
    const float* a0 = arow0 + c * KCHUNK;
    const float* a1 = arow1 + c * KCHUNK;
#pragma unroll 4
    for (int k = 0; k < KCHUNK; k += 4) {
      v2f aa0 = *(const v2f*)(a0 + k);
      v2f aa1 = *(const v2f*)(a1 + k);
#pragma unroll
      for (int nt = 0; nt < 4; ++nt) {
        v2f bb = *(const v2f*)(shp + (nt * 16 + lo) * KPITCH + k + 2 * hi);
        acc[nt]     = wmma4(aa0, bb, acc[nt]);
        acc[4 + nt] = wmma4(aa1, bb, acc[4 + nt]);
      }
    }
    __syncthreads();                            // before buffer reuse
  }
}

// ---------------------------------------------------------------------------
// Kernel 1: QKV projection, epilogue scatters to (B, H, S, DK).
// ---------------------------------------------------------------------------
__global__ __launch_bounds__(128)
void qkv_gemm_kernel(const float* __restrict__ x,
                     const float* __restrict__ wq,
                     const float* __restrict__ wk,
                     const float* __restrict__ wv,
                     float* __restrict__ Qo, float* __restrict__ Ko,
                     float* __restrict__ Vo)
{
  __shared__ float shw[2][64 * KPITCH];
  const int lane = threadIdx.x & 31;
  const int wid  = threadIdx.x >> 5;
  const int hi   = lane >> 4, lo = lane & 15;
  const int m0w = blockIdx.x * 128 + wid * 32;
  const int n0  = blockIdx.y * 64;

  const float* W;
  float* O;
  if (blockIdx.z == 0)      { W = wq; O = Qo; }
  else if (blockIdx.z == 1) { W = wk; O = Ko; }
  else                      { W = wv; O = Vo; }

  v8f acc[8] = {};
  gemm_core(x, W, shw, m0w, n0, wid, lane, acc);

#pragma unroll
  for (int mt = 0; mt < 2; ++mt)
#pragma unroll
    for (int nt = 0; nt < 4; ++nt) {
      const int n = n0 + nt * 16 + lo;
      const int h = n >> 6, d = n & 63;
#pragma unroll
      for (int r = 0; r < 8; ++r) {
        const int m = m0w + mt * 16 + r + 8 * hi;
        const int b = m >> 11, s = m & (SEQ - 1);
        O[(((size_t)(b * NHEAD + h) * SEQ + s) * DK) + d] = acc[mt * 4 + nt][r];
      }
    }
}

// ---------------------------------------------------------------------------
// Kernel 2: RoPE on Q and K (B,H,S,DK layout); folds 1/sqrt(DK) into Q.
// ---------------------------------------------------------------------------
__global__ void rope_kernel(float* __restrict__ Q, float* __restrict__ K,
                            const int* __restrict__ pos)
{
  const int idx = blockIdx.x * blockDim.x + threadIdx.x;
  const int total = BSZ * NHEAD * SEQ * (DK / 2);
  if (idx >= total) return;
  const int p = idx & 31;
  const int s = (idx >> 5) & (SEQ - 1);

  const float inv_freq = __powf(10000.0f, -(float)p / 32.0f);
  const float ang = (float)pos[s] * inv_freq;
  float sn, cs;
  __sincosf(ang, &sn, &cs);

  float2 q = *(float2*)(Q + (size_t)idx * 2);
  float2 k = *(float2*)(K + (size_t)idx * 2);
  const float qscale = 0.125f;
  float2 qo = make_float2((q.x * cs - q.y * sn) * qscale,
                          (q.x * sn + q.y * cs) * qscale);
  float2 ko = make_float2(k.x * cs - k.y * sn,
                          k.x * sn + k.y * cs);
  *(float2*)(Q + (size_t)idx * 2) = qo;
  *(float2*)(K + (size_t)idx * 2) = ko;
}

// ---------------------------------------------------------------------------
// Kernel 3: causal flash attention (one wave per (b,h, 16-query tile)).
// Q.K^T via WMMA; online softmax with 16-lane shfl_xor reductions (wave32);
// P transposed C-layout -> A-layout through LDS; O += P.V via WMMA.
// ---------------------------------------------------------------------------
__global__ __launch_bounds__(32)
void flash_attn_kernel(const float* __restrict__ Q, const float* __restrict__ K,
                       const float* __restrict__ V, float* __restrict__ AO)
{
  __shared__ float pbuf[16 * 16];
  const int lane = threadIdx.x & 31;
  const int hi = lane >> 4, lo = lane & 15;
  const int qt = blockIdx.x;
  const int bh = blockIdx.y;
  const int b = bh >> 4, h = bh & 15;

  const float* qbase = Q + ((size_t)bh * SEQ + qt * 16 + lo) * DK + 2 * hi;
  v2f qa[16];
#pragma unroll
  for (int t = 0; t < 16; ++t) qa[t] = *(const v2f*)(qbase + 4 * t);

  v8f oacc[4] = {};
  float m[8], l[8];
#pragma unroll
  for (int r = 0; r < 8; ++r) { m[r] = -__builtin_inff(); l[r] = 0.0f; }

  const float* krow = K + ((size_t)bh * SEQ + lo) * DK + 2 * hi;

  for (int kt = 0; kt <= qt; ++kt) {
    v8f sAcc = {};
    const float* kk = krow + (size_t)kt * 16 * DK;
#pragma unroll
    for (int t = 0; t < 16; ++t) {
      v2f kb = *(const v2f*)(kk + 4 * t);
      sAcc = wmma4(qa[t], kb, sAcc);
    }

    float sc[8];
#pragma unroll
    for (int r = 0; r < 8; ++r) {
      float v = sAcc[r];
      if (kt == qt && lo > (r + 8 * hi)) v = -__builtin_inff();
      float vm = v;
#pragma unroll
      for (int off = 1; off < 16; off <<= 1)
        vm = fmaxf(vm, __shfl_xor(vm, off, 32));
      const float mnew = fmaxf(m[r], vm);
      const float scl = __expf(m[r] - mnew);
      const float p = __expf(v - mnew);
      float ps = p;
#pragma unroll
      for (int off = 1; off < 16; off <<= 1)
        ps += __shfl_xor(ps, off, 32);
      l[r] = l[r] * scl + ps;
      m[r] = mnew;
      sc[r] = scl;
      pbuf[(r + 8 * hi) * 16 + lo] = p;
    }
#pragma unroll
    for (int nt = 0; nt < 4; ++nt)
#pragma unroll
      for (int r = 0; r < 8; ++r) oacc[nt][r] *= sc[r];
    __syncthreads();

    const float* vrow = V + ((size_t)bh * SEQ + kt * 16) * DK;
#pragma unroll
    for (int t = 0; t < 4; ++t) {
      v2f pa = *(const v2f*)(&pbuf[lo * 16 + 4 * t + 2 * hi]);
      const int kr = 4 * t + 2 * hi;
#pragma unroll
      for (int nt = 0; nt < 4; ++nt) {
        v2f vb = { vrow[(size_t)kr * DK + nt * 16 + lo],
                   vrow[(size_t)(kr + 1) * DK + nt * 16 + lo] };
        oacc[nt] = wmma4(pa, vb, oacc[nt]);
      }
    }
    __syncthreads();
  }

#pragma unroll
  for (int nt = 0; nt < 4; ++nt)
#pragma unroll
    for (int r = 0; r < 8; ++r) {
      const int s_idx = qt * 16 + r + 8 * hi;
      AO[(size_t)(b * SEQ + s_idx) * D_MODEL + h * DK + nt * 16 + lo] =
          oacc[nt][r] / l[r];
    }
}

// ---------------------------------------------------------------------------
// Kernel 4: output projection  out = AO @ wo^T (plain row-major store).
// ---------------------------------------------------------------------------
__global__ __launch_bounds__(128)
void out_gemm_kernel(const float* __restrict__ A, const float* __restrict__ W,
                     float* __restrict__ Out)
{
  __shared__ float shw[2][64 * KPITCH];
  const int lane = threadIdx.x & 31;
  const int wid  = threadIdx.x >> 5;
  const int hi   = lane >> 4, lo = lane & 15;
  const int m0w = blockIdx.x * 128 + wid * 32;
  const int n0  = blockIdx.y * 64;

  v8f acc[8] = {};
  gemm_core(A, W, shw, m0w, n0, wid, lane, acc);

#pragma unroll
  for (int mt = 0; mt < 2; ++mt)
#pragma unroll
    for (int nt = 0; nt < 4; ++nt)
#pragma unroll
      for (int r = 0; r < 8; ++r)
        Out[(size_t)(m0w + mt * 16 + r + 8 * hi) * D_MODEL + n0 + nt * 16 + lo] =
            acc[mt * 4 + nt][r];
}

// ---------------------------------------------------------------------------
extern "C" void kernel_launch(void* const* d_in, const int* in_sizes, int n_in,
                              void* d_out, int out_size, void* d_ws, size_t ws_size,
                              hipStream_t stream) {
  const float* x   = (const float*)d_in[0];
  const int*   pos = (const int*)  d_in[1];
  const float* wq  = (const float*)d_in[2];
  const float* wk  = (const float*)d_in[3];
  const float* wv  = (const float*)d_in[4];
  const float* wo  = (const float*)d_in[5];
  float* out = (float*)d_out;

  const size_t tensorElems = (size_t)BSZ * SEQ * D_MODEL;
  float* Q  = (float*)d_ws;
  float* K  = Q + tensorElems;
  float* V  = K + tensorElems;
  float* AO = V + tensorElems;

  // 1) QKV projections: blocks of 128 threads cover 128(M) x 64(N); z=Q/K/V
  dim3 g1(BSZ * SEQ / 128, D_MODEL / 64, 3), b1(128);
  qkv_gemm_kernel<<<g1, b1, 0, stream>>>(x, wq, wk, wv, Q, K, V);

  // 2) RoPE on Q,K
  const int pairs = BSZ * NHEAD * SEQ * (DK / 2);
  rope_kernel<<<(pairs + 255) / 256, 256, 0, stream>>>(Q, K, pos);

  // 3) causal flash attention
  dim3 g3(SEQ / 16, BSZ * NHEAD), b3(32);
  flash_attn_kernel<<<g3, b3, 0, stream>>>(Q, K, V, AO);

  // 4) output projection
  dim3 g4(BSZ * SEQ / 128, D_MODEL / 64), b4(128);
  out_gemm_kernel<<<g4, b4, 0, stream>>>(AO, wo, out);
}